// PostProcess_59897613910468
// MI455X (gfx1250) — compile-verified
//
#include <hip/hip_runtime.h>
#include <math.h>

typedef unsigned int u32;
typedef unsigned long long u64;
typedef __attribute__((ext_vector_type(4))) u32 u32x4;
typedef __attribute__((ext_vector_type(8))) u32 u32x8;
typedef __attribute__((ext_vector_type(2))) float v2f;
typedef __attribute__((ext_vector_type(8))) float v8f;

#define NB 32
#define NQ 900
#define NC 1203
#define QC (NQ * NC)            // 1082700 (divisible by 4)
#define NSEL 300
#define BINS 4096
#define CANDMAX 4096
#define TILE 4096               // elements per TDM tile (16 KB)
#define TILES_PER_BLK 16
#define CHUNK (TILE * TILES_PER_BLK)   // 65536 elements per block
#define HIST_BLKS ((QC + CHUNK - 1) / CHUNK)  // 17
#define IOU_THR 0.5f

#if __has_builtin(__builtin_amdgcn_wmma_f32_16x16x4_f32)
#define BOX_VIA_WMMA 1
#else
#define BOX_VIA_WMMA 0
#endif

// ---- order-preserving float -> u32 key (monotonic for IEEE754 totals) ----
__device__ __forceinline__ u32 mono_key(float f) {
  u32 b = __float_as_uint(f);
  return (b & 0x80000000u) ? ~b : (b | 0x80000000u);
}
__device__ __forceinline__ float inv_key(u32 k) {
  u32 b = (k & 0x80000000u) ? (k ^ 0x80000000u) : ~k;
  return __uint_as_float(b);
}

// ---- CDNA5 Tensor Data Mover: 1-D tile load, Global -> LDS -----------------
// D# group0 (4 SGPRs) + group1 (8 SGPRs) per CDNA5 ISA;
// VADDR2/3 disabled => tensor up to 2D. OOB elements zero-filled by TDM.
__device__ __forceinline__ void tdm_load_1d(u32 lds_addr, u64 gaddr,
                                            u32 tensor_elems, u32 tile_elems) {
  u32x4 g0;
  g0[0] = 1u;                                   // count=1, user desc
  g0[1] = lds_addr;                             // lds_addr (bytes)
  g0[2] = (u32)gaddr;                           // global_addr[31:0]
  g0[3] = ((u32)(gaddr >> 32) & 0x01FFFFFFu) | (2u << 30); // addr[56:32] | type=2
  u32x8 g1;
  g1[0] = (2u << 16);                           // workgroup_mask=0, data_size=4B
  g1[1] = (tensor_elems & 0xFFFFu) << 16;       // tensor_dim0[15:0]
  g1[2] = (tensor_elems >> 16) | (1u << 16);    // tensor_dim0[31:16] | tensor_dim1=1
  g1[3] = (tile_elems & 0xFFFFu) << 16;         // tile_dim0
  g1[4] = 1u;                                   // tile_dim1=1, tile_dim2=0
  g1[5] = tensor_elems;                         // tensor_dim0_stride[31:0]
  g1[6] = 0u;
  g1[7] = 0u;
  asm volatile("tensor_load_to_lds %0, %1" :: "s"(g0), "s"(g1) : "memory");
}

// ---------------------------------------------------------------------------
__global__ void zero_kernel(u32* __restrict__ p, int n) {
  int i = blockIdx.x * blockDim.x + threadIdx.x;
  if (i < n) p[i] = 0u;
}

// ---- Pass 1: per-batch 4096-bin histogram of monotonic keys ---------------
// grid = (HIST_BLKS, NB), block = 256 (8 waves). Wave 0 drives a double-
// buffered TDM pipeline; all waves histogram from LDS with ds atomics.
__global__ void __launch_bounds__(256) hist_kernel(const float* __restrict__ logits,
                                                   u32* __restrict__ hist) {
  __shared__ u32 lhist[BINS];                      // 16 KB
  __shared__ __align__(16) float tile[2][TILE];    // 2 x 16 KB
  const int b = blockIdx.y;
  const int tid = threadIdx.x;

  for (int i = tid; i < BINS; i += 256) lhist[i] = 0u;

  const long long chunk_start = (long long)blockIdx.x * CHUNK;
  long long remain = (long long)QC - chunk_start;
  int nt = 0;
  if (remain > 0) {
    nt = (int)((remain + TILE - 1) / TILE);
    if (nt > TILES_PER_BLK) nt = TILES_PER_BLK;
  }

  const u64 gbase = (u64)(const void*)(logits) + (u64)b * QC * 4ull;
  const u32 lds_lo = (u32)(u64)(const void*)&tile[0][0];
  const u32 lds_hi = (u32)(u64)(const void*)&tile[1][0];

  if (nt > 0 && tid == 0) {
    u64 ga = gbase + (u64)chunk_start * 4ull;
    tdm_load_1d(lds_lo, ga, (u32)remain, TILE);
  }

  for (int t = 0; t < nt; ++t) {
    if (tid == 0) {
      if (t + 1 < nt) {
        long long ts = chunk_start + (long long)(t + 1) * TILE;
        u64 ga = gbase + (u64)ts * 4ull;
        tdm_load_1d(((t + 1) & 1) ? lds_hi : lds_lo, ga, (u32)(QC - ts), TILE);
        __builtin_amdgcn_s_wait_tensorcnt(1);  // tile t done, t+1 in flight
      } else {
        __builtin_amdgcn_s_wait_tensorcnt(0);  // drain
      }
    }
    __syncthreads();                           // tile t visible to all waves

    const float* tb = tile[t & 1];
    const long long tstart = chunk_start + (long long)t * TILE;
#pragma unroll
    for (int k2 = 0; k2 < 4; ++k2) {
      const int e = k2 * 1024 + tid * 4;
      float4 v = *(const float4*)(tb + e);
      float a[4] = {v.x, v.y, v.z, v.w};
      const long long g = tstart + e;
#pragma unroll
      for (int k3 = 0; k3 < 4; ++k3) {
        if (g + k3 < (long long)QC) {
          atomicAdd(&lhist[mono_key(a[k3]) >> 20], 1u);
        }
      }
    }
    __syncthreads();                           // safe to overwrite buffer
  }

  __syncthreads();
  u32* gh = hist + (u64)b * BINS;
  for (int i = tid; i < BINS; i += 256) {
    u32 c = lhist[i];
    if (c) atomicAdd(&gh[i], c);
  }
}

// ---- Find threshold bin: largest bin T with count(bins >= T) >= NSEL -----
__global__ void scan_kernel(const u32* __restrict__ hist, u32* __restrict__ thr) {
  const int b = blockIdx.x;
  if (threadIdx.x != 0) return;
  const u32* h = hist + (u64)b * BINS;
  u32 acc = 0;
  int tb = 0;
  for (int i = BINS - 1; i >= 0; --i) {
    u32 c = h[i];
    if (acc + c >= NSEL) { tb = i; break; }
    acc += c;
  }
  thr[b] = (u32)tb;
}

// ---- Pass 2: compact candidates >= threshold bin --------------------------
__global__ void __launch_bounds__(256) compact_kernel(const float* __restrict__ logits,
                                                      const u32* __restrict__ thr,
                                                      u32* __restrict__ cnt,
                                                      u64* __restrict__ cand) {
  const int b = blockIdx.y;
  const u32 thrKey = thr[b] << 20;
  const float4* src = (const float4*)(logits + (u64)b * QC);
  const int nvec = QC / 4;
  const int stride = gridDim.x * blockDim.x;
  u64* cb = cand + (u64)b * CANDMAX;
  for (int v = blockIdx.x * blockDim.x + threadIdx.x; v < nvec; v += stride) {
    __builtin_prefetch((const void*)(src + v + stride), 0, 1);  // global_prefetch_b8
    float4 f = src[v];
    float a[4] = {f.x, f.y, f.z, f.w};
#pragma unroll
    for (int k = 0; k < 4; ++k) {
      u32 key = mono_key(a[k]);
      if (key >= thrKey) {
        u32 pos = atomicAdd(&cnt[b], 1u);
        if (pos < CANDMAX) {
          u32 idx = (u32)(4 * v + k);
          cb[pos] = ((u64)key << 32) | (u64)(u32)(~idx); // ~idx: ties -> lower idx first
        }
      }
    }
  }
}

// ---- Exact top-300 via bitonic sort; emit scores/labels/boxes -------------
// Box transform runs on the matrix core: [x0,y0,x1,y1] = [cx,cy,w,h] @ B4x4
// with B folding xyxy conversion + per-batch (iw,ih) scaling, executed as
// V_WMMA_F32_16X16X4_F32 (16 boxes per WMMA, 19 WMMAs per batch).
__global__ void __launch_bounds__(1024) select_kernel(const float* __restrict__ bbox,
                                                      const int* __restrict__ tsz,
                                                      const u32* __restrict__ cnt,
                                                      const u64* __restrict__ cand,
                                                      float* __restrict__ out) {
  __shared__ u64 arr[CANDMAX];                  // 32 KB
  const int b = blockIdx.x;
  const int tid = threadIdx.x;
  u32 n = cnt[b];
  if (n > CANDMAX) n = CANDMAX;
  const u64* cb = cand + (u64)b * CANDMAX;
  for (int i = tid; i < CANDMAX; i += 1024) arr[i] = (i < (int)n) ? cb[i] : 0ull;
  __syncthreads();

  // bitonic sort, descending
  for (u32 k = 2; k <= CANDMAX; k <<= 1) {
    for (u32 j = k >> 1; j > 0; j >>= 1) {
      for (u32 t = tid; t < CANDMAX / 2; t += 1024) {
        u32 i = ((t & ~(j - 1)) << 1) | (t & (j - 1));
        u32 p = i | j;
        u64 x = arr[i], y = arr[p];
        bool up = ((i & k) == 0);
        if (up ? (x < y) : (x > y)) { arr[i] = y; arr[p] = x; }
      }
      __syncthreads();
    }
  }

#if BOX_VIA_WMMA
  {
    const int lane = tid & 31;
    const int wv = tid >> 5;
    if (wv < (NSEL + 15) / 16) {                // 19 waves cover 304 >= 300 boxes
      const int wbase = wv * 16;
      const int nn = lane & 15;
      const bool hi = lane >= 16;
      // A: lanes 0-15 = (cx,cy) of box wbase+nn (K=0,1);
      //    lanes 16-31 = (w,h) of the same box  (K=2,3).
      u64 pk = arr[wbase + nn];
      u32 idx = ~((u32)pk);
      u32 qi = idx / NC;
      if (pk == 0ull || qi >= NQ) qi = 0;       // padding-safe
      const float* bp = bbox + ((u64)b * NQ + qi) * 4;
      float a0 = hi ? bp[2] : bp[0];
      float a1 = hi ? bp[3] : bp[1];
      // B: rows K0/K2 in VGPR0 (lo/hi lanes), rows K1/K3 in VGPR1
      const float ihh = (float)tsz[b * 2 + 0];
      const float iww = (float)tsz[b * 2 + 1];
      float row0 = (nn == 0 || nn == 2) ? iww : 0.f;                    // cx coeffs
      float row1 = (nn == 1 || nn == 3) ? ihh : 0.f;                    // cy coeffs
      float row2 = (nn == 0) ? -0.5f * iww : ((nn == 2) ? 0.5f * iww : 0.f); // w
      float row3 = (nn == 1) ? -0.5f * ihh : ((nn == 3) ? 0.5f * ihh : 0.f); // h
      v2f A = {a0, a1};
      v2f Bm = {hi ? row2 : row0, hi ? row3 : row1};
      v8f Cz = {};
      v8f D = __builtin_amdgcn_wmma_f32_16x16x4_f32(
          /*neg_a=*/false, A, /*neg_b=*/false, Bm,
          /*c_mod=*/(short)0, Cz, /*reuse_a=*/false, /*reuse_b=*/false);
      // D: lane n (n<4) holds coord n of rows m = r2 + 8*hi
      if (nn < 4) {
        float* obase = out + 2ull * NB * NSEL;
#pragma unroll
        for (int r2 = 0; r2 < 8; ++r2) {
          int box = wbase + r2 + (hi ? 8 : 0);
          if (box < NSEL) obase[((u64)b * NSEL + box) * 4 + nn] = D[r2];
        }
      }
    }
  }
#endif

  if (tid < NSEL) {
    u64 pk = arr[tid];
    float score = 0.f, labf = 0.f;
    if (pk != 0ull) {
      u32 key = (u32)(pk >> 32);
      u32 idx = ~((u32)pk);
      float logit = inv_key(key);
      score = 1.0f / (1.0f + expf(-logit));
      u32 qi = idx / NC;
      labf = (float)(idx - qi * NC);
#if !BOX_VIA_WMMA
      float4 bx = *(const float4*)(bbox + ((u64)b * NQ + qi) * 4);
      float ih = (float)tsz[b * 2 + 0];
      float iw = (float)tsz[b * 2 + 1];
      float* ob = out + 2ull * NB * NSEL + ((u64)b * NSEL + tid) * 4;
      ob[0] = (bx.x - 0.5f * bx.z) * iw;
      ob[1] = (bx.y - 0.5f * bx.w) * ih;
      ob[2] = (bx.x + 0.5f * bx.z) * iw;
      ob[3] = (bx.y + 0.5f * bx.w) * ih;
#endif
    }
    out[(u64)b * NSEL + tid] = score;
    out[(u64)NB * NSEL + (u64)b * NSEL + tid] = labf;
  }
}

// ---- Greedy sequential NMS (matches reference fori_loop exactly) ----------
__global__ void __launch_bounds__(320) nms_kernel(float* __restrict__ out) {
  __shared__ float4 sb[NSEL];
  __shared__ float sa[NSEL];
  __shared__ int kp[NSEL];
  const int b = blockIdx.x;
  const int tid = threadIdx.x;
  const float* boxes = out + 2ull * NB * NSEL + (u64)b * NSEL * 4;
  if (tid < NSEL) {
    float4 v = *(const float4*)(boxes + (u64)tid * 4);
    sb[tid] = v;
    sa[tid] = fmaxf(v.z - v.x, 0.f) * fmaxf(v.w - v.y, 0.f);
  }
  __syncthreads();
  for (int i = 0; i < NSEL; ++i) {
    int pred = 0;
    if (tid < i) {
      float4 bi = sb[i], bj = sb[tid];
      float lx = fmaxf(bi.x, bj.x), ly = fmaxf(bi.y, bj.y);
      float rx = fminf(bi.z, bj.z), ry = fminf(bi.w, bj.w);
      float inter = fmaxf(rx - lx, 0.f) * fmaxf(ry - ly, 0.f);
      float uni = sa[i] + sa[tid] - inter;
      float iou = inter / fmaxf(uni, 1e-9f);
      pred = (iou > IOU_THR) && kp[tid];
    }
    int sup = __syncthreads_or(pred);
    if (tid == 0) kp[i] = !sup;
    __syncthreads();
  }
  if (tid < NSEL) {
    out[6ull * NB * NSEL + (u64)b * NSEL + tid] = kp[tid] ? 1.0f : 0.0f;
  }
}

// ---------------------------------------------------------------------------
extern "C" void kernel_launch(void* const* d_in, const int* in_sizes, int n_in,
                              void* d_out, int out_size, void* d_ws, size_t ws_size,
                              hipStream_t stream) {
  const float* logits = (const float*)d_in[0];   // (32, 900, 1203) f32
  const float* bbox   = (const float*)d_in[1];   // (32, 900, 4)    f32
  const int*   tsz    = (const int*)d_in[2];     // (32, 2)         i32
  float* out = (float*)d_out;                    // 67200 f32

  // workspace layout
  u32* hist = (u32*)d_ws;                        // NB*BINS
  u32* thr  = hist + (u64)NB * BINS;             // NB
  u32* cnt  = thr + NB;                          // NB
  u64* cand = (u64*)(cnt + NB);                  // NB*CANDMAX u64 (8B aligned)

  const int zwords = NB * BINS + 2 * NB;
  zero_kernel<<<(zwords + 255) / 256, 256, 0, stream>>>(hist, zwords);
  hist_kernel<<<dim3(HIST_BLKS, NB), 256, 0, stream>>>(logits, hist);
  scan_kernel<<<NB, 32, 0, stream>>>(hist, thr);
  compact_kernel<<<dim3(64, NB), 256, 0, stream>>>(logits, thr, cnt, cand);
  select_kernel<<<NB, 1024, 0, stream>>>(bbox, tsz, cnt, cand, out);
  nms_kernel<<<NB, 320, 0, stream>>>(out);
}